// GCN_homo_38371237823055
// MI455X (gfx1250) — compile-verified
//
#include <hip/hip_runtime.h>
#include <math.h>

typedef __attribute__((ext_vector_type(2))) float v2f;
typedef __attribute__((ext_vector_type(8))) float v8f;

#define F_IN 512          // input feature dim (fixed by reference)
#define C_G 16            // GCN output channels
#define WAVES_PER_WG 8
#define MT 4              // 16-row tiles per wave (M=64 per wave)

// ---------------------------------------------------------------------------
// Kernel A: xw = x @ Wg  via V_WMMA_F32_16X16X4_F32
//   - Wg (512x16, 32 KB) staged in LDS pair-packed: lds[p][n] = (Wg[2p][n], Wg[2p+1][n])
//   - Each wave owns 4 M-tiles (64 rows) sharing one B fragment per K-step:
//       4x global_load_b64 (NT) + 1x ds_load_b64 + 4x v_wmma per K-step
//   - x loads are non-temporal: 205 MB read-once stream must not evict the
//     L2-resident graph-phase data (xw/agg/dinv/edge_index).
// ---------------------------------------------------------------------------
__global__ __launch_bounds__(256)
void gemm_xw_kernel(const float* __restrict__ x, const float* __restrict__ Wg,
                    float* __restrict__ xw, int N) {
    __shared__ float2 ldsB[F_IN / 2][C_G];   // 32 KB

    const int tid = threadIdx.x;

    // Cooperative stage of Wg into LDS (pair-packed along K)
    for (int idx = tid; idx < (F_IN / 2) * C_G; idx += 256) {
        const int p = idx >> 4;          // K-pair index
        const int n = idx & 15;          // output channel
        float2 v;
        v.x = Wg[(2 * p) * C_G + n];
        v.y = Wg[(2 * p + 1) * C_G + n];
        ldsB[p][n] = v;
    }
    __syncthreads();

    const int wave = tid >> 5;
    const int lane = tid & 31;
    const int half = lane >> 4;          // 0: K=k,k+1   1: K=k+2,k+3
    const int l16  = lane & 15;

    const long waveIdx = (long)blockIdx.x * WAVES_PER_WG + wave;
    const long row0    = waveIdx * (MT * 16);      // first row of this wave's 64-row block
    if (row0 >= N) return;                          // wave-uniform: EXEC all-ones below

    // Per-tile A base pointers (rows clamped; stores are guarded in the tail)
    const float* aBase[MT];
    #pragma unroll
    for (int t = 0; t < MT; ++t) {
        long r = row0 + t * 16 + l16;
        if (r >= N) r = N - 1;
        aBase[t] = x + r * F_IN + 2 * half;
    }

    v8f acc[MT] = {};
    #pragma unroll 2
    for (int k = 0; k < F_IN; k += 4) {
        const float2 bq = ldsB[(k >> 1) + half][l16];   // B 4x16 fragment
        v2f b;
        b[0] = bq.x;
        b[1] = bq.y;
        #pragma unroll
        for (int t = 0; t < MT; ++t) {
            const v2f a = __builtin_nontemporal_load((const v2f*)(aBase[t] + k));
            acc[t] = __builtin_amdgcn_wmma_f32_16x16x4_f32(
                /*neg_a=*/false, a, /*neg_b=*/false, b,
                /*c_mod=*/(short)0, acc[t], /*reuse_a=*/false, /*reuse_b=*/false);
        }
    }

    // C/D layout: VGPR r, lanes 0-15 -> M=r, lanes 16-31 -> M=r+8 ; col = lane&15
    #pragma unroll
    for (int t = 0; t < MT; ++t) {
        const long tb = row0 + t * 16;
        if (tb + 16 <= N) {              // full tile (always, except clamped tail)
            #pragma unroll
            for (int r = 0; r < 8; ++r)
                xw[(tb + r + 8 * half) * C_G + l16] = acc[t][r];
        } else if (tb < N) {             // partial tail tile
            #pragma unroll
            for (int r = 0; r < 8; ++r) {
                const long orow = tb + r + 8 * half;
                if (orow < N) xw[orow * C_G + l16] = acc[t][r];
            }
        }
    }
}

// ---------------------------------------------------------------------------
// Degree / normalization kernels
// ---------------------------------------------------------------------------
__global__ void init_deg_kernel(float* __restrict__ deg, int N) {
    const int i = blockIdx.x * 256 + threadIdx.x;
    if (i < N) deg[i] = 1.0f;            // self-loop contribution
}

__global__ void deg_kernel(const long long* __restrict__ ei,
                           float* __restrict__ deg, int E) {
    const int e = blockIdx.x * 256 + threadIdx.x;
    if (e < E) {
        const int col = (int)ei[(long)E + e];   // target index
        atomicAdd(&deg[col], 1.0f);
    }
}

// dinv[i] = deg^-0.5 ; agg[i][c] = xw[i][c] * dinv[i]^2   (self-loop term)
__global__ void dinv_self_kernel(const float* __restrict__ xw,
                                 const float* __restrict__ deg,
                                 float* __restrict__ dinv,
                                 float* __restrict__ agg, int N) {
    const long g = (long)blockIdx.x * 256 + threadIdx.x;
    const long i = g >> 4;
    const int  c = (int)(g & 15);
    if (i < N) {
        const float d  = deg[i];         // always >= 1 (self loops)
        const float di = rsqrtf(d);
        if (c == 0) dinv[i] = di;
        agg[i * C_G + c] = xw[i * C_G + c] * (di * di);
    }
}

// ---------------------------------------------------------------------------
// Edge scatter: agg[col] += xw[row] * dinv[row]*dinv[col]   (16 lanes / edge)
// agg (6.4 MB) and dinv (0.4 MB) are L2-resident -> atomics resolve in L2.
// ---------------------------------------------------------------------------
__global__ void scatter_kernel(const long long* __restrict__ ei,
                               const float* __restrict__ xw,
                               const float* __restrict__ dinv,
                               float* __restrict__ agg, int E) {
    const long g = (long)blockIdx.x * 256 + threadIdx.x;
    const long e = g >> 4;
    const int  c = (int)(g & 15);
    if (e < E) {
        const int row = (int)ei[e];            // source (x_j)
        const int col = (int)ei[(long)E + e];  // target
        const float norm = dinv[row] * dinv[col];
        atomicAdd(&agg[(long)col * C_G + c], xw[(long)row * C_G + c] * norm);
    }
}

// ---------------------------------------------------------------------------
// Fused MLP head: relu(agg+bg) -> relu(@W1+b1) -> relu(@W2+b2) -> sigmoid(@W3+b3)
// One thread per node; ~800 FMAs, weights are uniform and L2/L0 cached.
// ---------------------------------------------------------------------------
__global__ __launch_bounds__(256)
void mlp_kernel(const float* __restrict__ agg,
                const float* __restrict__ bg,
                const float* __restrict__ W1, const float* __restrict__ b1,
                const float* __restrict__ W2, const float* __restrict__ b2,
                const float* __restrict__ W3, const float* __restrict__ b3,
                float* __restrict__ out, int N) {
    const int i = blockIdx.x * 256 + threadIdx.x;
    if (i >= N) return;

    float h1[16];
    #pragma unroll
    for (int c = 0; c < 16; ++c)
        h1[c] = fmaxf(agg[(long)i * C_G + c] + bg[c], 0.0f);

    float h2[16];
    #pragma unroll
    for (int j = 0; j < 16; ++j) {
        float s = b1[j];
        #pragma unroll
        for (int c = 0; c < 16; ++c) s = fmaf(h1[c], W1[c * 16 + j], s);
        h2[j] = fmaxf(s, 0.0f);
    }

    float h3[32];
    #pragma unroll
    for (int j = 0; j < 32; ++j) {
        float s = b2[j];
        #pragma unroll
        for (int c = 0; c < 16; ++c) s = fmaf(h2[c], W2[c * 32 + j], s);
        h3[j] = fmaxf(s, 0.0f);
    }

    float s = b3[0];
    #pragma unroll
    for (int c = 0; c < 32; ++c) s = fmaf(h3[c], W3[c], s);

    out[i] = 1.0f / (1.0f + __expf(-s));
}

// ---------------------------------------------------------------------------
// Launch
// ---------------------------------------------------------------------------
extern "C" void kernel_launch(void* const* d_in, const int* in_sizes, int n_in,
                              void* d_out, int out_size, void* d_ws, size_t ws_size,
                              hipStream_t stream) {
    const float*     x  = (const float*)d_in[0];
    const long long* ei = (const long long*)d_in[1];   // int64 [2, E]
    const float*     Wg = (const float*)d_in[2];
    const float*     bg = (const float*)d_in[3];
    const float*     W1 = (const float*)d_in[4];
    const float*     b1 = (const float*)d_in[5];
    const float*     W2 = (const float*)d_in[6];
    const float*     b2 = (const float*)d_in[7];
    const float*     W3 = (const float*)d_in[8];
    const float*     b3 = (const float*)d_in[9];
    float* out = (float*)d_out;

    const int N = in_sizes[0] / F_IN;   // 100000
    const int E = in_sizes[1] / 2;      // 3200000

    // Workspace layout (all fp32): xw[N*16] | agg[N*16] | deg[N] | dinv[N]
    char* ws = (char*)d_ws;
    float* xw   = (float*)(ws);
    float* agg  = (float*)(ws + (size_t)N * C_G * 4);
    float* deg  = (float*)(ws + (size_t)N * C_G * 4 * 2);
    float* dinv = (float*)(ws + (size_t)N * C_G * 4 * 2 + (size_t)N * 4);

    // 1) xw = x @ Wg  (WMMA f32, 64 rows per wave)
    const long tiles  = (N + 15) / 16;
    const long wavesA = (tiles + MT - 1) / MT;
    const int  wgA    = (int)((wavesA + WAVES_PER_WG - 1) / WAVES_PER_WG);
    gemm_xw_kernel<<<wgA, 256, 0, stream>>>(x, Wg, xw, N);

    // 2) degrees (self-loop + in-edges)
    init_deg_kernel<<<(N + 255) / 256, 256, 0, stream>>>(deg, N);
    deg_kernel<<<(E + 255) / 256, 256, 0, stream>>>(ei, deg, E);

    // 3) dinv + self-loop init of agg
    {
        const long work = (long)N * C_G;
        dinv_self_kernel<<<(int)((work + 255) / 256), 256, 0, stream>>>(xw, deg, dinv, agg, N);
    }

    // 4) edge scatter with symmetric normalization
    {
        const long work = (long)E * C_G;
        scatter_kernel<<<(int)((work + 255) / 256), 256, 0, stream>>>(ei, xw, dinv, agg, E);
    }

    // 5) fused MLP head + sigmoid
    mlp_kernel<<<(N + 255) / 256, 256, 0, stream>>>(agg, bg, W1, b1, W2, b2, W3, b3, out, N);
}